// Model_39676907886975
// MI455X (gfx1250) — compile-verified
//
#include <hip/hip_runtime.h>

typedef __attribute__((ext_vector_type(2))) float v2f;
typedef __attribute__((ext_vector_type(8))) float v8f;

#define SEQ 16
#define DIM 64
#define WAVES_PER_BLOCK 8
#define LDS_STRIDE 18   // padded row stride (floats): 72B, 8B-aligned, conflict-free halves

__device__ __forceinline__ v8f wmma_f32(v2f a, v2f b, v8f c) {
  return __builtin_amdgcn_wmma_f32_16x16x4_f32(false, a, false, b, (short)0, c,
                                               false, false);
}

// threefry2x32, key = (0, 42)  (jax.random.key(42)), 20 rounds
__device__ __forceinline__ void threefry2x32_k42(unsigned c0, unsigned c1,
                                                 unsigned& o0, unsigned& o1) {
  const unsigned ks0 = 0u, ks1 = 42u;
  const unsigned ks2 = ks0 ^ ks1 ^ 0x1BD11BDAu;
  const unsigned ks[3] = {ks0, ks1, ks2};
  unsigned x0 = c0 + ks0;
  unsigned x1 = c1 + ks1;
  const int rotA[4] = {13, 15, 26, 6};
  const int rotB[4] = {17, 29, 16, 24};
#pragma unroll
  for (int g = 0; g < 5; ++g) {
#pragma unroll
    for (int r = 0; r < 4; ++r) {
      const int rr = (g & 1) ? rotB[r] : rotA[r];
      x0 += x1;
      x1 = (x1 << rr) | (x1 >> (32 - rr));
      x1 ^= x0;
    }
    x0 += ks[(g + 1) % 3];
    x1 += ks[(g + 2) % 3] + (unsigned)(g + 1);
  }
  o0 = x0;
  o1 = x1;
}

__device__ __forceinline__ float tf_uniform(unsigned bits) {
  return __uint_as_float((bits >> 9) | 0x3f800000u) - 1.0f;
}

// GEMM1: attn_raw(16x16) = Q(16x64) x K^T(64x16), 16 chained f32 WMMAs.
__device__ __forceinline__ v8f gemm_qkT(const float* __restrict__ qb,
                                        const float* __restrict__ kb,
                                        int p, int hi) {
  v8f acc = {};
#pragma unroll
  for (int j = 0; j < 16; ++j) {
    const int col = 4 * j + 2 * hi;
    v2f a = *(const v2f*)(qb + p * DIM + col);  // A: Q[p][col..col+1]
    v2f b = *(const v2f*)(kb + p * DIM + col);  // B: K^T[kd][n] = K[n][kd]
    acc = wmma_f32(a, b, acc);
  }
  return acc;
}

// exp (no max-subtraction needed: |score| bounded, fp32-safe; softmax invariant
// to the +ones(16,16) row-constant mask) then C-layout -> A-layout via LDS.
__device__ __forceinline__ void exp_transpose(v8f acc, float* __restrict__ wl,
                                              int p, int hi, v2f (&ach)[4]) {
  const float SCALE = 0.35355339059327373f;  // 1/sqrt(8)
#pragma unroll
  for (int i = 0; i < 8; ++i)
    wl[(i + 8 * hi) * LDS_STRIDE + p] = __expf(acc[i] * SCALE);
  // DS ops are in-order within a wave; compiler inserts the dscnt wait.
#pragma unroll
  for (int j = 0; j < 4; ++j)
    ach[j] = *(const v2f*)(wl + p * LDS_STRIDE + 4 * j + 2 * hi);
}

// Row sums via WMMA with B = ones: D[m][n] = sum_t A[m][t] for every n.
// C-layout puts s[r+8*hi] in VGPR r -- exactly the layout of the GEMM2 output,
// so normalization is deferred to a post-GEMM2 row scale.
__device__ __forceinline__ v8f row_sums(const v2f (&ach)[4]) {
  const v2f ones = {1.0f, 1.0f};
  v8f s = {};
#pragma unroll
  for (int j = 0; j < 4; ++j) s = wmma_f32(ach[j], ones, s);
  return s;
}

// GEMM2: out = (A_unnorm x V) * rcp(rowsum), 4 N-tiles of 16.
__device__ __forceinline__ void gemm_av_store(const v2f (&ach)[4], v8f sums,
                                              const float* __restrict__ vb,
                                              float* __restrict__ ob, int p, int hi) {
  float rsinv[8];
#pragma unroll
  for (int r = 0; r < 8; ++r) rsinv[r] = __builtin_amdgcn_rcpf(sums[r]);
#pragma unroll
  for (int nt = 0; nt < 4; ++nt) {
    v8f o = {};
#pragma unroll
    for (int j = 0; j < 4; ++j) {
      const int r0 = 4 * j + 2 * hi;
      v2f bv;                                  // each b32 load = 2 full 64B lines
      bv.x = vb[r0 * DIM + 16 * nt + p];
      bv.y = vb[(r0 + 1) * DIM + 16 * nt + p];
      o = wmma_f32(ach[j], bv, o);
    }
#pragma unroll
    for (int r = 0; r < 8; ++r)
      ob[(r + 8 * hi) * DIM + 16 * nt + p] = o[r] * rsinv[r];
  }
}

// One wave handles the tile pair (t, t+4096): JAX's threefry counter scheme
// pairs flat index q (< 2^20) with q + 2^20, which is the same (s,t) position
// of the partner tile -- one threefry eval feeds both tiles' dropout.
__global__ __launch_bounds__(256) void attn_sdp_kernel(
    const float* __restrict__ q, const float* __restrict__ k,
    const float* __restrict__ v, float* __restrict__ out, int n_pairs) {
  __shared__ float lds_attn[WAVES_PER_BLOCK][2][SEQ * LDS_STRIDE];

  const int lane = threadIdx.x & 31;
  const int wave = threadIdx.x >> 5;
  const int pair = blockIdx.x * WAVES_PER_BLOCK + wave;

  const int p  = lane & 15;
  const int hi = lane >> 4;

  const size_t baseA = (size_t)pair * (SEQ * DIM);
  const size_t baseB = (size_t)(pair + n_pairs) * (SEQ * DIM);
  const float* __restrict__ vA = v + baseA;
  const float* __restrict__ vB = v + baseB;

  // Warm caches for V while we chew on the QK^T chains.
  __builtin_prefetch(vA + lane * 16, 0, 3);
  __builtin_prefetch(vA + 512 + lane * 16, 0, 3);
  __builtin_prefetch(vB + lane * 16, 0, 3);
  __builtin_prefetch(vB + 512 + lane * 16, 0, 3);

  v8f accA = gemm_qkT(q + baseA, k + baseA, p, hi);
  v8f accB = gemm_qkT(q + baseB, k + baseB, p, hi);

  v2f achA[4], achB[4];
  exp_transpose(accA, &lds_attn[wave][0][0], p, hi, achA);
  exp_transpose(accB, &lds_attn[wave][1][0], p, hi, achB);

  const v8f sumA = row_sums(achA);   // pre-dropout denominators
  const v8f sumB = row_sums(achB);

  // Dropout p=0.3 in A-layout: lane (p,hi) chunk j holds attn[p][4j+2hi+{0,1}].
  const unsigned HALF = 1u << 20;            // (256*32*16*16)/2
  const unsigned tbase = (unsigned)pair * 256u + (unsigned)(p * 16 + 2 * hi);
  const float KEEP = 1.4285714285714286f;    // 1/(1-p)
#pragma unroll
  for (int j = 0; j < 4; ++j) {
#pragma unroll
    for (int c = 0; c < 2; ++c) {
      const unsigned flat = tbase + (unsigned)(4 * j + c);  // tileA flat < 2^20
      unsigned w0, w1;
      threefry2x32_k42(flat, flat + HALF, w0, w1);
      const float kA = (tf_uniform(w0) < 0.7f) ? KEEP : 0.0f;  // tileA: word0
      const float kB = (tf_uniform(w1) < 0.7f) ? KEEP : 0.0f;  // tileB: word1
      if (c == 0) { achA[j].x *= kA; achB[j].x *= kB; }
      else        { achA[j].y *= kA; achB[j].y *= kB; }
    }
  }

  gemm_av_store(achA, sumA, vA, out + baseA, p, hi);
  gemm_av_store(achB, sumB, vB, out + baseB, p, hi);
}

extern "C" void kernel_launch(void* const* d_in, const int* in_sizes, int n_in,
                              void* d_out, int out_size, void* d_ws, size_t ws_size,
                              hipStream_t stream) {
  const float* q = (const float*)d_in[0];
  const float* k = (const float*)d_in[1];
  const float* v = (const float*)d_in[2];
  float* out = (float*)d_out;

  const int n_tiles = in_sizes[0] / (SEQ * DIM);        // B*H = 8192
  const int n_pairs = n_tiles / 2;                      // 4096
  const int n_blocks = n_pairs / WAVES_PER_BLOCK;       // 512
  attn_sdp_kernel<<<dim3(n_blocks), dim3(256), 0, stream>>>(q, k, v, out, n_pairs);
}